// GATBlock_2113123909822
// MI455X (gfx1250) — compile-verified
//
#include <hip/hip_runtime.h>

#define NN 50000
#define EE 1600000
#define INF_ 128
#define HH 4
#define DD 32
#define HD 128           // HH*DD
#define NEG_SLOPE 0.2f

typedef __attribute__((ext_vector_type(2))) float v2f;
typedef __attribute__((ext_vector_type(8))) float v8f;

// ---- monotonic float<->uint encoding for atomicMax-based segment_max ----
__device__ __forceinline__ unsigned int ford(float f) {
  unsigned int b = __float_as_uint(f);
  return (b & 0x80000000u) ? ~b : (b | 0x80000000u);
}
__device__ __forceinline__ float funord(unsigned int u) {
  unsigned int b = (u & 0x80000000u) ? (u ^ 0x80000000u) : ~u;
  return __uint_as_float(b);
}
#define FORD_NEG_INF 0x007FFFFFu   // ford(-inf)

// ---------------- init kernels ----------------
__global__ void k_zero_f32(float* __restrict__ p, int n) {
  int i = blockIdx.x * blockDim.x + threadIdx.x;
  if (i < n) p[i] = 0.0f;
}
__global__ void k_init_ms(unsigned int* __restrict__ m, float* __restrict__ s, int n) {
  int i = blockIdx.x * blockDim.x + threadIdx.x;
  if (i < n) { m[i] = FORD_NEG_INF; s[i] = 0.0f; }
}

// ---------------- projection: h = feat @ W  (fp32 WMMA 16x16x4) ----------------
// grid.x = NN/16 row tiles; 256 threads = 8 waves; wave w owns col tile w*16.
__global__ __launch_bounds__(256) void k_proj(const float* __restrict__ feat,
                                              const float* __restrict__ W,
                                              float* __restrict__ h) {
  __shared__ float As[16 * 132];        // 16x128 A tile, stride 132 (bank-conflict pad)
  const int m0 = blockIdx.x * 16;
  const int t  = threadIdx.x;

  // cooperative stage of the 16x128 feature tile (8 floats per thread, 2x float4)
  {
    const int row = t >> 4;             // 0..15
    const int col = (t & 15) * 8;       // 0..120
    const float4* gp = (const float4*)(feat + (size_t)(m0 + row) * INF_ + col);
    float4 x0 = gp[0];
    float4 x1 = gp[1];
    float* lp = As + row * 132 + col;
    *(float4*)(lp)     = x0;
    *(float4*)(lp + 4) = x1;
  }
  __syncthreads();

  const int lane  = t & 31;
  const int wave  = t >> 5;
  const int n0    = wave * 16;          // output col tile
  const int r     = lane & 15;          // A row / B col within tile
  const int khalf = (lane >> 4) << 1;   // lanes 0-15 -> K{0,1}; lanes 16-31 -> K{2,3}

  v8f acc = {};
  for (int k0 = 0; k0 < INF_; k0 += 4) {
    const int kk = k0 + khalf;
    v2f a;
    a.x = As[r * 132 + kk];
    a.y = As[r * 132 + kk + 1];
    v2f b;
    b.x = W[(size_t)(kk)     * HD + n0 + r];
    b.y = W[(size_t)(kk + 1) * HD + n0 + r];
    acc = __builtin_amdgcn_wmma_f32_16x16x4_f32(false, a, false, b,
                                                (short)0, acc, false, false);
  }

  // C/D layout: VGPR j, lanes 0-15 -> M=j, lanes 16-31 -> M=j+8; N = lane&15
  const int mh = (lane >> 4) * 8;
#pragma unroll
  for (int j = 0; j < 8; ++j) {
    h[(size_t)(m0 + j + mh) * HD + n0 + r] = acc[j];
  }
}

// ---------------- el/er: per-(node,head) dot with al/ar ----------------
__global__ void k_elr(const float* __restrict__ h, const float* __restrict__ al,
                      const float* __restrict__ ar, float* __restrict__ el,
                      float* __restrict__ er) {
  int tid = blockIdx.x * blockDim.x + threadIdx.x;
  if (tid >= NN * HH) return;
  const int n  = tid / HH;
  const int hh = tid % HH;
  const float* hp  = h  + (size_t)n * HD + hh * DD;
  const float* alp = al + hh * DD;
  const float* arp = ar + hh * DD;
  float sl = 0.0f, sr = 0.0f;
#pragma unroll
  for (int d = 0; d < DD; d += 4) {
    float4 hv = *(const float4*)(hp + d);
    float4 av = *(const float4*)(alp + d);
    float4 rv = *(const float4*)(arp + d);
    sl += hv.x * av.x + hv.y * av.y + hv.z * av.z + hv.w * av.w;
    sr += hv.x * rv.x + hv.y * rv.y + hv.z * rv.z + hv.w * rv.w;
  }
  el[tid] = sl;
  er[tid] = sr;
}

// ---------------- edge pass 1: leaky-relu logits + segment max ----------------
__global__ void k_edge_logit(const int* __restrict__ src, const int* __restrict__ dst,
                             const float* __restrict__ el, const float* __restrict__ er,
                             float* __restrict__ ebuf, unsigned int* __restrict__ m) {
  int tid = blockIdx.x * blockDim.x + threadIdx.x;
  if (tid >= EE * HH) return;
  const int e  = tid >> 2;       // HH == 4
  const int hh = tid & 3;
  const int sd = src[e], dd = dst[e];
  float x = el[sd * HH + hh] + er[dd * HH + hh];
  x = (x > 0.0f) ? x : NEG_SLOPE * x;
  ebuf[tid] = x;
  atomicMax(&m[dd * HH + hh], ford(x));
}

// ---------------- edge pass 2: exp(e - max) + segment sum ----------------
__global__ void k_edge_exp(const int* __restrict__ dst, const unsigned int* __restrict__ m,
                           float* __restrict__ ebuf, float* __restrict__ s) {
  int tid = blockIdx.x * blockDim.x + threadIdx.x;
  if (tid >= EE * HH) return;
  const int e  = tid >> 2;
  const int hh = tid & 3;
  const int dd = dst[e];
  const float mm = funord(m[dd * HH + hh]);  // finite: dst has >=1 edge (this one)
  const float ex = __expf(ebuf[tid] - mm);
  ebuf[tid] = ex;
  atomicAdd(&s[dd * HH + hh], ex);
}

// ---------------- edge pass 3: alpha * h[src] scatter-add ----------------
// one thread per (edge, head, 4-wide chunk of D): EE*HH*8 threads
__global__ __launch_bounds__(256) void k_edge_msg(const int* __restrict__ src,
                                                  const int* __restrict__ dst,
                                                  const float* __restrict__ ebuf,
                                                  const float* __restrict__ s,
                                                  const float* __restrict__ h,
                                                  float* __restrict__ acc) {
  int tid = blockIdx.x * blockDim.x + threadIdx.x;
  if (tid >= EE * HH * 8) return;
  const int e   = tid >> 5;          // HH*8 == 32 threads per edge
  const int rem = tid & 31;
  const int hh  = rem >> 3;
  const int q   = rem & 7;           // 4-float chunk within head
  const int sd  = src[e], dd = dst[e];
  const float alpha = ebuf[e * HH + hh] / s[dd * HH + hh];
  const float4 hv = *(const float4*)(h + (size_t)sd * HD + hh * DD + q * 4);
  float* ap = acc + (size_t)dd * HD + hh * DD + q * 4;
  atomicAdd(ap + 0, alpha * hv.x);
  atomicAdd(ap + 1, alpha * hv.y);
  atomicAdd(ap + 2, alpha * hv.z);
  atomicAdd(ap + 3, alpha * hv.w);
}

// ---------------- head mean ----------------
__global__ void k_mean(const float* __restrict__ acc, float* __restrict__ out) {
  int tid = blockIdx.x * blockDim.x + threadIdx.x;
  if (tid >= NN * DD) return;
  const int n = tid >> 5;            // DD == 32
  const int d = tid & 31;
  const float* a = acc + (size_t)n * HD + d;
  out[tid] = 0.25f * (a[0] + a[32] + a[64] + a[96]);
}

extern "C" void kernel_launch(void* const* d_in, const int* in_sizes, int n_in,
                              void* d_out, int out_size, void* d_ws, size_t ws_size,
                              hipStream_t stream) {
  const float* feat = (const float*)d_in[0];
  const int*   src_g[2] = { (const int*)d_in[1], (const int*)d_in[3] };
  const int*   dst_g[2] = { (const int*)d_in[2], (const int*)d_in[4] };
  const float* W_g[2]   = { (const float*)d_in[5], (const float*)d_in[8] };
  const float* al_g[2]  = { (const float*)d_in[6], (const float*)d_in[9] };
  const float* ar_g[2]  = { (const float*)d_in[7], (const float*)d_in[10] };
  float* out = (float*)d_out;

  // workspace layout (floats); h/m/s/ebuf reused across graphs (serial stream)
  float*        acc  = (float*)d_ws;                        // NN*HD
  float*        hbuf = acc  + (size_t)NN * HD;              // NN*HD
  float*        el   = hbuf + (size_t)NN * HD;              // NN*HH
  float*        er   = el   + (size_t)NN * HH;              // NN*HH
  unsigned int* m    = (unsigned int*)(er + (size_t)NN * HH); // NN*HH
  float*        s    = (float*)(m + (size_t)NN * HH);       // NN*HH
  float*        ebuf = s + (size_t)NN * HH;                 // EE*HH

  const int T = 256;
  k_zero_f32<<<(NN * HD + T - 1) / T, T, 0, stream>>>(acc, NN * HD);

  for (int g = 0; g < 2; ++g) {
    k_proj<<<NN / 16, T, 0, stream>>>(feat, W_g[g], hbuf);
    k_elr<<<(NN * HH + T - 1) / T, T, 0, stream>>>(hbuf, al_g[g], ar_g[g], el, er);
    k_init_ms<<<(NN * HH + T - 1) / T, T, 0, stream>>>(m, s, NN * HH);
    k_edge_logit<<<(EE * HH + T - 1) / T, T, 0, stream>>>(src_g[g], dst_g[g], el, er, ebuf, m);
    k_edge_exp<<<(EE * HH + T - 1) / T, T, 0, stream>>>(dst_g[g], m, ebuf, s);
    k_edge_msg<<<(EE * HH * 8 + T - 1) / T, T, 0, stream>>>(src_g[g], dst_g[g], ebuf, s, hbuf, acc);
  }

  k_mean<<<(NN * DD + T - 1) / T, T, 0, stream>>>(acc, out);
}